// DistanceAttention_27951647163022
// MI455X (gfx1250) — compile-verified
//
#include <hip/hip_runtime.h>
#include <hip/hip_bf16.h>

// ---------------------------------------------------------------------------
// DistanceAttention on MI455X (gfx1250), f32 WMMA path.
//
//   y'[b,j,:] = exp(mask[b,j]/e) * (x[b,j,:] @ W^T + b_in)        (proj kernel)
//   out[b,i,:] = sum_j exp(-|i-j|/e) * y'[b,j,:]                  (attn kernel)
//
// exp(-|i-j|/e) underflows to exactly 0.0f (f32, incl. denormals) for
// |i-j| >= ~289, so the 4096x4096 attn matrix is exactly banded with
// half-bandwidth BAND=320.  Banded WMMA does ~5.5 GFLOP instead of 34.4.
// Everything (35 MB) is L2-resident (192 MB L2).
// ---------------------------------------------------------------------------

typedef __attribute__((ext_vector_type(2))) float v2f;
typedef __attribute__((ext_vector_type(8))) float v8f;

#define TT   4096
#define DD   256
#define BB   4
#define BAND 320                    // multiple of 16; covers all nonzero f32 weights
#define INV_E 0.36787944117144233f  // 1/e

// D = A(16x4) * B(4x16) + C(16x16), all f32.
// 8-arg pattern per bridge doc: (neg_a, A, neg_b, B, c_mod, C, reuse_a, reuse_b)
__device__ __forceinline__ v8f wmma_f32(v2f a, v2f b, v8f c) {
  return __builtin_amdgcn_wmma_f32_16x16x4_f32(false, a, false, b,
                                               (short)0, c, false, false);
}

// ---------------------------------------------------------------------------
// proj: y'[m, n] = exp(mask[m]/e) * (sum_k x[m,k] * W[n,k] + bias[n])
//   M = B*T = 16384, N = D = 256, K = D = 256.
// One wave (32 thr) per (16-row M-tile) x (64-col N-quad).
// A-matrix layout (ISA 7.12.2, 32-bit 16x4): lane l<16 -> M=l, VGPR0=K0,VGPR1=K1;
// lanes 16-31 -> M=l-16, VGPR0=K2, VGPR1=K3.  B (4x16) assumed symmetric to C:
// VGPR p, lane-half h -> row K = 2h+p, col N = lane%16.
// ---------------------------------------------------------------------------
__global__ __launch_bounds__(32)
void da_proj_kernel(const float* __restrict__ x, const float* __restrict__ mask,
                    const float* __restrict__ W, const float* __restrict__ bias,
                    float* __restrict__ y) {
  const int lane = threadIdx.x & 31;
  const int l = lane & 15;
  const int h = lane >> 4;
  const int m0 = blockIdx.x * 16;       // 0..16383 step 16
  const int N0 = blockIdx.y * 64;       // 0..255 step 64

  v8f zero = {};
  v8f c0 = zero, c1 = zero, c2 = zero, c3 = zero;

  const float* xrow = x + (m0 + l) * DD;   // A: row m0+l, adjacent K pair -> float2
  const float* w0 = W + (N0 +  0 + l) * DD;
  const float* w1 = W + (N0 + 16 + l) * DD;
  const float* w2 = W + (N0 + 32 + l) * DD;
  const float* w3 = W + (N0 + 48 + l) * DD;

  for (int k0 = 0; k0 < DD; k0 += 4) {
    const int kk = k0 + 2 * h;           // this lane's K pair start
    v2f a = *(const v2f*)(xrow + kk);    // (K=kk, K=kk+1) -> VGPR0, VGPR1
    v2f b0 = *(const v2f*)(w0 + kk);     // B^T element pair W[n, kk..kk+1]
    v2f b1 = *(const v2f*)(w1 + kk);
    v2f b2 = *(const v2f*)(w2 + kk);
    v2f b3 = *(const v2f*)(w3 + kk);
    c0 = wmma_f32(a, b0, c0);
    c1 = wmma_f32(a, b1, c1);
    c2 = wmma_f32(a, b2, c2);
    c3 = wmma_f32(a, b3, c3);
  }

  // C layout: VGPR r -> rows r (lanes 0-15) / r+8 (lanes 16-31), col = N + l.
  // mask flat index == global row index (mask is [B,1,T] -> [B*T]).
#pragma unroll
  for (int r = 0; r < 8; ++r) {
    const int row = m0 + r + 8 * h;
    const float mf = __expf(mask[row] * INV_E);
    float* yr = y + row * DD;
    yr[N0 +  0 + l] = (c0[r] + bias[N0 +  0 + l]) * mf;
    yr[N0 + 16 + l] = (c1[r] + bias[N0 + 16 + l]) * mf;
    yr[N0 + 32 + l] = (c2[r] + bias[N0 + 32 + l]) * mf;
    yr[N0 + 48 + l] = (c3[r] + bias[N0 + 48 + l]) * mf;
  }
}

// ---------------------------------------------------------------------------
// attn: out[b, I0+m, n] = sum_{j in band} exp(-|I0+m - j|/e) * y'[b, j, n]
// Block = 256 threads = 8 waves; one block per (batch, 16-row tile).
// Wave w owns 32 output columns [32w, 32w+32): two 16x16 C tiles.
// A tile is generated analytically: 2 v_exp_f32 per lane per K-step,
// co-executing on the trans pipe alongside the WMMAs.
// ---------------------------------------------------------------------------
__global__ __launch_bounds__(256)
void da_attn_kernel(const float* __restrict__ y, float* __restrict__ out) {
  const int lane = threadIdx.x & 31;
  const int l = lane & 15;
  const int h = lane >> 4;
  const int wv = threadIdx.x >> 5;                // 0..7
  const int it = blockIdx.x & (TT / 16 - 1);      // row tile within batch
  const int bi = blockIdx.x >> 8;                 // batch (T/16 = 256 tiles)
  const int I0 = it * 16;
  const int N0 = wv * 32;

  const float* yb = y + (bi * TT) * DD;

  int jlo = I0 - BAND; if (jlo < 0) jlo = 0;      // multiples of 16 -> clean steps of 4
  int jhi = I0 + 16 + BAND; if (jhi > TT) jhi = TT;

  v8f zero = {};
  v8f c0 = zero, c1 = zero;
  const float m = (float)(I0 + l);                // this lane's output row (A-matrix M)

  for (int j0 = jlo; j0 < jhi; j0 += 4) {
    const int kk = j0 + 2 * h;                    // this lane's K pair (columns j)
    const float k0f = (float)kk;
    v2f a;
    a.x = __expf(-fabsf(m - k0f) * INV_E);
    a.y = __expf(-fabsf(m - (k0f + 1.0f)) * INV_E);

    const float* r0 = yb + kk * DD;               // y' row j = kk   (B K-row 2h+0)
    const float* r1 = r0 + DD;                    // y' row j = kk+1 (B K-row 2h+1)
    v2f b0, b1;
    b0.x = r0[N0 + l];       b0.y = r1[N0 + l];
    b1.x = r0[N0 + 16 + l];  b1.y = r1[N0 + 16 + l];

    c0 = wmma_f32(a, b0, c0);
    c1 = wmma_f32(a, b1, c1);
  }

  float* ob = out + (bi * TT + I0) * DD;
#pragma unroll
  for (int r = 0; r < 8; ++r) {
    const int row = r + 8 * h;
    ob[row * DD + N0 + l]      = c0[r];
    ob[row * DD + N0 + 16 + l] = c1[r];
  }
}

// ---------------------------------------------------------------------------
extern "C" void kernel_launch(void* const* d_in, const int* in_sizes, int n_in,
                              void* d_out, int out_size, void* d_ws, size_t ws_size,
                              hipStream_t stream) {
  const float* x    = (const float*)d_in[0];   // [4,4096,256]
  const float* mask = (const float*)d_in[1];   // [4,1,4096]
  const float* W    = (const float*)d_in[2];   // [256,256]
  const float* bias = (const float*)d_in[3];   // [256]
  float* out = (float*)d_out;                  // [4,4096,256]
  float* y   = (float*)d_ws;                   // scratch: 16384*256 floats = 16.8 MB

  // proj: 1024 M-tiles x 4 N-quads, 1 wave each
  dim3 gproj(BB * TT / 16, DD / 64);
  da_proj_kernel<<<gproj, 32, 0, stream>>>(x, mask, W, bias, y);

  // attn: one 8-wave block per (batch, 16-row tile)
  da_attn_kernel<<<BB * TT / 16, 256, 0, stream>>>(y, out);
}